// ContrastiveLoss_50044958933555
// MI455X (gfx1250) — compile-verified
//
#include <hip/hip_runtime.h>

typedef __attribute__((ext_vector_type(2))) float v2f;
typedef __attribute__((ext_vector_type(8))) float v8f;
typedef __attribute__((ext_vector_type(4))) int v4i;
typedef __attribute__((address_space(1))) v4i gv4i;   // global int4
typedef __attribute__((address_space(3))) v4i lv4i;   // LDS int4

namespace {
constexpr int NB = 64, NR = 36, NW = 40, ND = 1024;
constexpr int KC = 64;     // K-chunk (floats) staged in LDS per iteration
constexpr int LSTR = 68;   // LDS row stride in floats (68%64==4 -> conflict-free b64 reads)
constexpr int NQ = KC / 4;            // float4 per row per chunk (16)
constexpr int STAGE_ELEMS = 48 * NQ;  // float4 per matrix per chunk (768)
constexpr float EPSF = 1e-8f;
constexpr float MARGIN = 0.2f;
constexpr float LSM_LSE = 6.0f;   // i2t softmax temperature
constexpr float LSM_AVG = 9.0f;   // t2i softmax temperature
constexpr float LLSE    = 6.0f;   // LSE aggregation temperature

// workspace layout (float offsets)
constexpr size_t WS_SCORES = 0;                                  // NB*NB
constexpr size_t WS_GS     = WS_SCORES + (size_t)NB * NB;        // NB*NW*NW
constexpr size_t WS_NS     = WS_GS + (size_t)NB * NW * NW;       // NB*NW
constexpr size_t WS_GI     = WS_NS + (size_t)NB * NW;            // NB*NR*NR
constexpr size_t WS_NIM    = WS_GI + (size_t)NB * NR * NR;       // NB*NR
}

#if __has_builtin(__builtin_amdgcn_global_load_async_to_lds_b128)
#define SCAN_ASYNC_LDS 1
#else
#define SCAN_ASYNC_LDS 0
#endif

__device__ __forceinline__ void async_cp16(const float* src, float* dst) {
#if SCAN_ASYNC_LDS
  __builtin_amdgcn_global_load_async_to_lds_b128((gv4i*)src, (lv4i*)dst, 0, 0);
#else
  (void)src; (void)dst;
#endif
}

__device__ __forceinline__ void async_wait0() {
#if SCAN_ASYNC_LDS
#if __has_builtin(__builtin_amdgcn_s_wait_asynccnt)
  __builtin_amdgcn_s_wait_asynccnt(0);
#else
  asm volatile("s_wait_asynccnt 0x0" ::: "memory");
#endif
#endif
}

__device__ __forceinline__ float lrelu(float x) { return x > 0.f ? x : 0.1f * x; }

__device__ __forceinline__ v8f wmma_f32_k4(v2f a, v2f b, v8f c) {
  // D = A(16x4) * B(4x16) + C, fp32 throughout
  return __builtin_amdgcn_wmma_f32_16x16x4_f32(false, a, false, b, (short)0, c,
                                               false, false);
}

struct StageBuf {  // double-buffered A/B chunk tiles
  float A0[48 * LSTR], B0[48 * LSTR], A1[48 * LSTR], B1[48 * LSTR];
};

// 48x48 (3x3 tiles of 16x16, one tile per wave, 9 waves / 288 threads) product
//   rows(aBase) x rows(bBase)^T  over K = ND, accumulated in f32 WMMA.
// Out-of-range rows/cols are index-clamped (NOT zeroed): they only pollute
// C rows >= an / cols >= bn, which are never read.
// A/B staged through double-buffered LDS chunks; threads 0..255 own 6 float4
// transfers each (2*48*16 = 1536 = 256*6), so global latency of chunk k+1 is
// hidden behind the wmma consume of chunk k.
__device__ __forceinline__ void wmma_gemm48_lds(
    const float* __restrict__ aBase, int an,
    const float* __restrict__ bBase, int bn,
    StageBuf* sbuf, float* ldsG, int tid) {
  const int lane = tid & 31, wave = tid >> 5;
  const int tw = wave / 3, tr = wave % 3;   // tile row (M blocks), tile col (N blocks)
  const int ln = lane & 15, hi = lane >> 4; // entity index within tile, K-half select
  const int aoff = (tw * 16 + ln) * LSTR + 2 * hi;
  const int boff = (tr * 16 + ln) * LSTR + 2 * hi;

  // per-thread staging descriptors (threads 0..255 active)
  const bool stager = tid < 256;
  const float* src6[6];
  int dof6[6];
  bool isB6[6];
  #pragma unroll
  for (int c = 0; c < 6; ++c) {
    const int idx = c * 256 + tid;                 // 0..1535
    const bool isB = idx >= STAGE_ELEMS;
    const int t = isB ? idx - STAGE_ELEMS : idx;
    const int row = t >> 4;                        // 0..47
    const int q = t & 15;                          // float4 slot in chunk
    const int n = isB ? bn : an;
    const int srow = row < n ? row : n - 1;        // clamp (garbage rows never read)
    src6[c] = (isB ? bBase : aBase) + (size_t)srow * ND + q * 4;
    dof6[c] = row * LSTR + q * 4;
    isB6[c] = isB;
  }

  float4 pre[6];  // register prefetch (sync fallback only)

  auto issue = [&](int kb, float* Ad, float* Bd) {
    if (stager) {
      #pragma unroll
      for (int c = 0; c < 6; ++c) {
#if SCAN_ASYNC_LDS
        async_cp16(src6[c] + kb, (isB6[c] ? Bd : Ad) + dof6[c]);
#else
        pre[c] = *reinterpret_cast<const float4*>(src6[c] + kb);
#endif
      }
    }
    (void)Ad; (void)Bd;
  };
  auto commit = [&](float* Ad, float* Bd) {
#if SCAN_ASYNC_LDS
    async_wait0();
    (void)Ad; (void)Bd;
#else
    if (stager) {
      #pragma unroll
      for (int c = 0; c < 6; ++c)
        *reinterpret_cast<float4*>((isB6[c] ? Bd : Ad) + dof6[c]) = pre[c];
    }
#endif
  };

  issue(0, sbuf->A0, sbuf->B0);
  commit(sbuf->A0, sbuf->B0);
  __syncthreads();

  v8f acc = {};
  for (int kb = 0; kb < ND; kb += KC) {
    const bool p1 = ((kb / KC) & 1) != 0;
    float* Ac = p1 ? sbuf->A1 : sbuf->A0;
    float* Bc = p1 ? sbuf->B1 : sbuf->B0;
    float* An = p1 ? sbuf->A0 : sbuf->A1;
    float* Bn = p1 ? sbuf->B0 : sbuf->B1;
    const bool more = kb + KC < ND;
    if (more) issue(kb + KC, An, Bn);     // overlap next-chunk fills with wmma
    #pragma unroll
    for (int kk = 0; kk < KC; kk += 4) {
      v2f a = *reinterpret_cast<const v2f*>(Ac + aoff + kk);
      v2f b = *reinterpret_cast<const v2f*>(Bc + boff + kk);
      acc = wmma_f32_k4(a, b, acc);
    }
    if (more) commit(An, Bn);
    __syncthreads();
  }
  // 16x16 f32 C/D layout: VGPR v -> row v (lanes 0-15) / row v+8 (lanes 16-31)
  #pragma unroll
  for (int v = 0; v < 8; ++v)
    ldsG[(tw * 16 + v + 8 * hi) * 48 + tr * 16 + ln] = acc[v];
}

// ---------------- K0: per-image / per-caption Gram matrices + norms ----------
__global__ __launch_bounds__(288) void scan_gram_kernel(
    const float* __restrict__ im, const float* __restrict__ s,
    float* __restrict__ ws) {
  __shared__ __align__(16) StageBuf sbuf;
  __shared__ float Gt[48 * 48];
  const int tid = threadIdx.x;
  const int b = blockIdx.x;
  if (b < NB) {                                   // Gram_s[j] = s_j s_j^T (40x40)
    const int j = b;
    const float* base = s + (size_t)j * NW * ND;
    wmma_gemm48_lds(base, NW, base, NW, &sbuf, Gt, tid);
    __syncthreads();
    float* outG = ws + WS_GS + (size_t)j * NW * NW;
    for (int idx = tid; idx < NW * NW; idx += 288)
      outG[idx] = Gt[(idx / NW) * 48 + (idx % NW)];
    if (tid < NW) ws[WS_NS + (size_t)j * NW + tid] = sqrtf(Gt[tid * 48 + tid]);
  } else {                                        // Gram_im[i] = im_i im_i^T (36x36)
    const int i = b - NB;
    const float* base = im + (size_t)i * NR * ND;
    wmma_gemm48_lds(base, NR, base, NR, &sbuf, Gt, tid);
    __syncthreads();
    float* outG = ws + WS_GI + (size_t)i * NR * NR;
    for (int idx = tid; idx < NR * NR; idx += 288)
      outG[idx] = Gt[(idx / NR) * 48 + (idx % NR)];
    if (tid < NR) ws[WS_NIM + (size_t)i * NR + tid] = sqrtf(Gt[tid * 48 + tid]);
  }
}

// post-GEMM working set; aliases the stage buffers (used strictly after them)
struct PostBuf {
  float GS[NW * NW];       // Gram_s[j]
  float GI[NR * NR];       // Gram_im[i]
  float AT1[NR * NW];      // i2t attention [r][w]
  float AT2[NW * NR];      // t2i attention [w][r]
  float RN[NW], WN[NR], NSn[NW], NIn[NR];
  float S1[NR], S2[NW], C1[NW], C2[NR];
};
union ShUnion { StageBuf st; PostBuf po; };

// ---------------- K1: one (image i, caption j) pair per workgroup ------------
__global__ __launch_bounds__(288) void scan_pair_kernel(
    const float* __restrict__ im, const float* __restrict__ s,
    const int* __restrict__ s_l, float* __restrict__ ws,
    float* __restrict__ out) {
  __shared__ __align__(16) ShUnion u;
  __shared__ float G[48 * 48];        // raw dot products G[w][r] = s_w . im_r
  const int tid = threadIdx.x;
  const int i = blockIdx.x, j = blockIdx.y;
  const float* sb = s  + (size_t)j * NW * ND;
  const float* ib = im + (size_t)i * NR * ND;

  // G = s_j (40xD) x im_i^T (Dx36)  via WMMA, padded to 48x48
  wmma_gemm48_lds(sb, NW, ib, NR, &u.st, G, tid);
  __syncthreads();   // stage buffers dead from here; u.po live

  float* GS = u.po.GS;  float* GI = u.po.GI;
  float* AT1 = u.po.AT1; float* AT2 = u.po.AT2;
  float* RN = u.po.RN;  float* WN = u.po.WN;
  float* NSn = u.po.NSn; float* NIn = u.po.NIn;
  float* S1 = u.po.S1;  float* S2 = u.po.S2;
  float* C1 = u.po.C1;  float* C2 = u.po.C2;

  // stage pair-invariant tables into LDS
  for (int idx = tid; idx < NW * NW; idx += 288) GS[idx] = ws[WS_GS + (size_t)j * NW * NW + idx];
  for (int idx = tid; idx < NR * NR; idx += 288) GI[idx] = ws[WS_GI + (size_t)i * NR * NR + idx];
  if (tid < NW) NSn[tid] = ws[WS_NS + (size_t)j * NW + tid];
  if (tid < NR) NIn[tid] = ws[WS_NIM + (size_t)i * NR + tid];

  int len = s_l[j]; if (len < 1) len = 1; if (len > NW) len = NW;

  // l2 norms of leaky(G): RN[w] over regions (i2t), WN[r] over valid words (t2i)
  if (tid < NW) {
    float acc = 0.f;
    for (int r = 0; r < NR; ++r) { float l = lrelu(G[tid * 48 + r]); acc += l * l; }
    RN[tid] = sqrtf(acc);
  } else if (tid >= 64 && tid < 64 + NR) {
    int r = tid - 64; float acc = 0.f;
    for (int w = 0; w < len; ++w) { float l = lrelu(G[w * 48 + r]); acc += l * l; }
    WN[r] = sqrtf(acc);
  }
  __syncthreads();

  if (tid < NR) {
    // ---- i2t: thread per region r; softmax over valid words ----
    const int r = tid;
    float mx = -3.4e38f;
    for (int w = 0; w < len; ++w)
      mx = fmaxf(mx, LSM_LSE * lrelu(G[w * 48 + r]) / (RN[w] + EPSF));
    float sum = 0.f;
    for (int w = 0; w < NW; ++w) {
      float e = 0.f;
      if (w < len) {
        float f = LSM_LSE * lrelu(G[w * 48 + r]) / (RN[w] + EPSF);
        e = expf(f - mx); sum += e;
      }
      AT1[r * NW + w] = e;
    }
    float inv = 1.f / sum, num = 0.f;
    for (int w = 0; w < len; ++w) {
      float aw = AT1[r * NW + w] * inv;
      AT1[r * NW + w] = aw;
      num += aw * G[w * 48 + r];               // im_r . wc_t[r]
    }
    float q = 0.f;                              // |wc_t[r]|^2 = a^T Gram_s a
    for (int w = 0; w < len; ++w) {
      float t = 0.f;
      for (int w2 = 0; w2 < len; ++w2) t += AT1[r * NW + w2] * GS[w * NW + w2];
      q += AT1[r * NW + w] * t;
    }
    S1[r] = num / fmaxf(NIn[r] * sqrtf(q), EPSF);
  } else if (tid >= 64 && tid < 64 + NW) {
    // ---- t2i: thread per word w; softmax over regions ----
    const int w = tid - 64;
    if (w < len) {
      float mx = -3.4e38f;
      for (int r = 0; r < NR; ++r)
        mx = fmaxf(mx, LSM_AVG * lrelu(G[w * 48 + r]) / (WN[r] + EPSF));
      float sum = 0.f;
      for (int r = 0; r < NR; ++r) {
        float f = LSM_AVG * lrelu(G[w * 48 + r]) / (WN[r] + EPSF);
        float e = expf(f - mx); AT2[w * NR + r] = e; sum += e;
      }
      float inv = 1.f / sum, num = 0.f;
      for (int r = 0; r < NR; ++r) {
        float ar = AT2[w * NR + r] * inv;
        AT2[w * NR + r] = ar;
        num += ar * G[w * 48 + r];             // s_w . wc_i[w]
      }
      float q = 0.f;                            // |wc_i[w]|^2 = a^T Gram_im a
      for (int r = 0; r < NR; ++r) {
        float t = 0.f;
        for (int r2 = 0; r2 < NR; ++r2) t += AT2[w * NR + r2] * GI[r * NR + r2];
        q += AT2[w * NR + r] * t;
      }
      S2[w] = num / fmaxf(NSn[w] * sqrtf(q), EPSF);
    } else {
      for (int r = 0; r < NR; ++r) AT2[w * NR + r] = 0.f;
      S2[w] = 0.f;
    }
  }
  __syncthreads();

  if (tid == 0) {
    float e1 = 0.f;
    for (int r = 0; r < NR; ++r) e1 += expf(LLSE * S1[r]);
    float e2 = 0.f;
    for (int w = 0; w < len; ++w) e2 += expf(LLSE * S2[w]);
    ws[WS_SCORES + (size_t)i * NB + j] = 0.5f * (logf(e1) / LLSE + logf(e2) / LLSE);
  }

  if (i == j) {  // diagonal attention outputs (block-uniform branch)
    if (tid < NW) {
      float c = 0.f;
      for (int r = 0; r < NR; ++r) c += AT1[r * NW + tid];
      C1[tid] = c;
    } else if (tid >= 64 && tid < 64 + NR) {
      int r = tid - 64; float c = 0.f;
      for (int w = 0; w < len; ++w) c += AT2[w * NR + r];
      C2[r] = c;
    }
    __syncthreads();
    const float invlen = 1.f / (float)len;
    for (int d = tid; d < ND; d += 288) {
      float at = 0.f;
      for (int w = 0; w < NW; ++w) at += C1[w] * sb[(size_t)w * ND + d];
      out[1 + (size_t)i * ND + d] = at * (1.f / (float)NR);
      float ai = 0.f;
      for (int r = 0; r < NR; ++r) ai += C2[r] * ib[(size_t)r * ND + d];
      out[1 + (size_t)NB * ND + (size_t)i * ND + d] = ai * invlen;
    }
  }
}

// ---------------- K2: hinge loss over the 64x64 score matrix -----------------
__global__ __launch_bounds__(64) void scan_loss_kernel(const float* __restrict__ ws,
                                                       float* __restrict__ out) {
  __shared__ float S[NB * NB];
  __shared__ float red[NB];
  const int t = threadIdx.x;
  for (int idx = t; idx < NB * NB; idx += 64) S[idx] = ws[WS_SCORES + idx];
  __syncthreads();
  const float dt = S[t * NB + t];
  float rm = 0.f, cm = 0.f;
  for (int k = 0; k < NB; ++k) {
    if (k != t) {
      rm = fmaxf(rm, fmaxf(0.f, MARGIN + S[t * NB + k] - dt)); // cost_s, row t
      cm = fmaxf(cm, fmaxf(0.f, MARGIN + S[k * NB + t] - dt)); // cost_im, col t
    }
  }
  red[t] = rm + cm;
  __syncthreads();
  if (t == 0) {
    float tot = 0.f;
    for (int k = 0; k < NB; ++k) tot += red[k];
    out[0] = tot;
  }
}

extern "C" void kernel_launch(void* const* d_in, const int* in_sizes, int n_in,
                              void* d_out, int out_size, void* d_ws, size_t ws_size,
                              hipStream_t stream) {
  const float* im  = (const float*)d_in[0];   // (64, 36, 1024) f32
  const float* s   = (const float*)d_in[1];   // (64, 40, 1024) f32
  const int*   s_l = (const int*)d_in[2];     // (64,) int32
  float* out = (float*)d_out;                 // [loss, attn_t(64*1024), attn_i(64*1024)]
  float* ws  = (float*)d_ws;

  scan_gram_kernel<<<dim3(2 * NB), 288, 0, stream>>>(im, s, ws);
  scan_pair_kernel<<<dim3(NB, NB), 288, 0, stream>>>(im, s, s_l, ws, out);
  scan_loss_kernel<<<1, 64, 0, stream>>>(ws, out);
}